// PlanarPhysORD_84224308675143
// MI455X (gfx1250) — compile-verified
//
#include <hip/hip_runtime.h>

// CDNA5 / gfx1250: wave32, FP32 WMMA 16x16x4 for the 10->64->3 MLP force model,
// state held in VGPRs across the 32-step scan, weights staged to LDS once and
// held as WMMA fragments in VGPRs. Fully branchless step body (EXEC stays all-1s,
// as WMMA requires) using native exp2/rcp/sqrt/sin/cos. Running pointers avoid
// per-step 64-bit address remultiplication.

typedef float v2f __attribute__((ext_vector_type(2)));
typedef float v8f __attribute__((ext_vector_type(8)));

#define PH      0.1f
#define MMASS   900.0f
#define JINER   1000.0f
#define PCIRC   2.0f
#define PC      0.5f

constexpr int D     = 17;   // state dim (xy2, th1, v2, om1, sk4, rpm4, uk3)
constexpr int STEPS = 32;
constexpr int BS    = 65536;
constexpr int KPAD  = 12;   // 10 input features padded to 12 (3 K-chunks of 4)
constexpr int W1S   = 68;   // LDS row stride (floats) for W1   -> conflict-free
constexpr int W2S   = 20;   // LDS row stride (floats) for W2^T
constexpr int HS    = 68;   // LDS row stride (floats) for hid  -> conflict-free

// Branchless tanh: 1 - 2/(exp2(2*log2e*x)+1). v_exp_f32 + v_rcp_f32, clamp
// keeps exp2 in range (saturates to +/-1 exactly where it should).
__device__ __forceinline__ float fast_tanhf(float x) {
    float k = x * 2.885390082f;                 // 2*log2(e)
    k = fminf(fmaxf(k, -36.0f), 36.0f);         // v_med3_num_f32
    const float e = __builtin_amdgcn_exp2f(k);  // v_exp_f32 (TRANS)
    return 1.0f - 2.0f * __builtin_amdgcn_rcpf(e + 1.0f); // v_rcp_f32 (TRANS)
}

__global__ __launch_bounds__(128) void PlanarPhysORD_wmma_kernel(
    const float* __restrict__ traj, const float* __restrict__ W1,
    const float* __restrict__ B1,   const float* __restrict__ W2,
    const float* __restrict__ B2,   float* __restrict__ out)
{
    __shared__ float w1s[KPAD * W1S];      //  3264 B
    __shared__ float w2s[64 * W2S];        //  5120 B
    __shared__ float hidls[4 * 32 * HS];   // 34816 B (per-wave private slabs)

    const int  tid    = threadIdx.x;
    const int  wav    = tid >> 5;
    const int  lane   = tid & 31;
    const int  lo     = lane & 15;
    const bool hiHalf = lane >= 16;
    const int  row    = blockIdx.x * 128 + wav * 32 + lane; // this lane's batch row
    float* hidb = &hidls[wav * 32 * HS];

    // ---- stage weights into LDS (zero-padded) ----
    for (int i = tid; i < KPAD * W1S; i += 128) w1s[i] = 0.0f;
    for (int i = tid; i < 64 * W2S;  i += 128) w2s[i] = 0.0f;
    __syncthreads();
    for (int i = tid; i < 10 * 64; i += 128) w1s[(i / 64) * W1S + (i % 64)] = W1[i];
    for (int i = tid; i < 64 * 3;  i += 128) w2s[(i / 3) * W2S + (i % 3)]  = W2[i]; // W2^T
    __syncthreads();

    // ---- step-invariant WMMA fragments held in VGPRs ----
    // Layer-1 B (W1): K-chunk c (k=4c..4c+3), N-tile t (cols 16t..16t+15).
    // B layout 4x16: v0 = row k0 (lanes 0-15) / row k2 (lanes 16-31); v1 = k1/k3.
    v2f bfr1[3][4];
#pragma unroll
    for (int c = 0; c < 3; ++c) {
        const int k0 = 4 * c + (hiHalf ? 2 : 0);
#pragma unroll
        for (int t = 0; t < 4; ++t) {
            bfr1[c][t].x = w1s[(k0    ) * W1S + lo + 16 * t];
            bfr1[c][t].y = w1s[(k0 + 1) * W1S + lo + 16 * t];
        }
    }
    // Layer-2 A (W2^T, 16x64 padded): A layout 16x4: v0 = col k0/k2, v1 = k1/k3.
    v2f afr2[16];
#pragma unroll
    for (int c = 0; c < 16; ++c) {
        const int k0 = 4 * c + (hiHalf ? 2 : 0);
        afr2[c].x = w2s[(k0    ) * W2S + lo];
        afr2[c].y = w2s[(k0 + 1) * W2S + lo];
    }
    float bias1[4];
#pragma unroll
    for (int t = 0; t < 4; ++t) bias1[t] = B1[lo + 16 * t];
    const float b2x = B2[0], b2y = B2[1], b2r = B2[2];

    // ---- load x0, emit out[0] ----
    float st[D];
    {
        const float* p = traj + (size_t)row * D;
        float*       o = out  + (size_t)row * D;
#pragma unroll
        for (int c = 0; c < D; ++c) { st[c] = p[c]; o[c] = st[c]; }
    }

    // Running pointers: one 64-bit add per step instead of mul-based rebuild.
    const size_t stride    = (size_t)BS * D;
    const float* ctrlp     = traj + stride + (size_t)row * D + (D - 3);
    float*       outp      = out  + stride + (size_t)row * D;

    // ---- 32 sequential steps (fully branchless body) ----
    for (int s = 0; s < STEPS; ++s) {
        // issue next control loads early so they overlap the MLP
        const float u0n = ctrlp[0], u1n = ctrlp[1], u2n = ctrlp[2];
        // prefetch the step after that (global_prefetch_b8); clamp to stay in bounds
        __builtin_prefetch(ctrlp + (s + 1 < STEPS ? stride : 0), 0, 1);

        const float vx = st[3], vy = st[4], om = st[5];
        const float vsum = __builtin_amdgcn_sqrtf(vx * vx + vy * vy); // raw v_sqrt_f32
        float fin[KPAD];
        fin[0] = vx; fin[1] = vy; fin[2] = om;
        fin[3] = st[14]; fin[4] = st[15]; fin[5] = st[16];
        fin[6] = vsum - st[10] * (PCIRC / 60.0f);
        fin[7] = vsum - st[11] * (PCIRC / 60.0f);
        fin[8] = vsum - st[12] * (PCIRC / 60.0f);
        fin[9] = vsum - st[13] * (PCIRC / 60.0f);
        fin[10] = 0.0f; fin[11] = 0.0f;

        // Layer-1 A fragments: M-tile m covers batch rows in lanes m*16..m*16+15.
        v2f afr1[2][3];
#pragma unroll
        for (int m = 0; m < 2; ++m) {
            const int src = lo | (m << 4);
#pragma unroll
            for (int c = 0; c < 3; ++c) {
                const float t0 = __shfl(fin[4 * c + 0], src, 32);
                const float t1 = __shfl(fin[4 * c + 1], src, 32);
                const float t2 = __shfl(fin[4 * c + 2], src, 32);
                const float t3 = __shfl(fin[4 * c + 3], src, 32);
                afr1[m][c].x = hiHalf ? t2 : t0;
                afr1[m][c].y = hiHalf ? t3 : t1;
            }
        }

        // Layer 1: hid = tanh(f_in @ W1 + b1); C layout: elem(M=r+8*hi, N=lo+16t).
#pragma unroll
        for (int m = 0; m < 2; ++m) {
#pragma unroll
            for (int t = 0; t < 4; ++t) {
                v8f acc = {};
#pragma unroll
                for (int c = 0; c < 3; ++c)
                    acc = __builtin_amdgcn_wmma_f32_16x16x4_f32(
                        false, afr1[m][c], false, bfr1[c][t], (short)0, acc,
                        false, false);
#pragma unroll
                for (int r = 0; r < 8; ++r) {
                    const int hrow = r + (hiHalf ? 8 : 0) + 16 * m;
                    hidb[hrow * HS + lo + 16 * t] = fast_tanhf(acc[r] + bias1[t]);
                }
            }
        }

        // Layer 2 (transposed): D = W2^T (16x64) x hid^T (64x16-batch) per N-tile.
        v8f acc2[2];
#pragma unroll
        for (int t = 0; t < 2; ++t) {
            v8f a = {};
#pragma unroll
            for (int c = 0; c < 16; ++c) {
                const int k0 = 4 * c + (hiHalf ? 2 : 0);
                const float2 hv =
                    *(const float2*)&hidb[(16 * t + lo) * HS + k0]; // ds_load_b64
                v2f bf; bf.x = hv.x; bf.y = hv.y;
                a = __builtin_amdgcn_wmma_f32_16x16x4_f32(
                    false, afr2[c], false, bf, (short)0, a, false, false);
            }
            acc2[t] = a;
        }
        // forces for this lane's row: rows m_out=0..2 live in vgprs 0..2, lanes 0-15.
        const float s0 = __shfl(acc2[1][0], lo, 32);
        const float s1 = __shfl(acc2[1][1], lo, 32);
        const float s2 = __shfl(acc2[1][2], lo, 32);
        const float fXx = (hiHalf ? s0 : acc2[0][0]) + b2x;
        const float fXy = (hiHalf ? s1 : acc2[0][1]) + b2y;
        const float fR  = (hiHalf ? s2 : acc2[0][2]) + b2r;

        // ---- semi-implicit planar integration (branchless, native sin/cos) ----
        const float fxm = (PC * PH) * fXx,          fym = (PC * PH) * fXy;
        const float fxp = ((1.0f - PC) * PH) * fXx, fyp = ((1.0f - PC) * PH) * fXy;
        const float fRm = (PC * PH) * fR,           fRp = ((1.0f - PC) * PH) * fR;
        const float ct = __cosf(st[2]), sn = __sinf(st[2]);
        const float fmwx = ct * fxm - sn * fym;
        const float fmwy = sn * fxm + ct * fym;
        const float xn  = st[0] + PH * vx + (PH / MMASS) * fmwx;
        const float yn  = st[1] + PH * vy + (PH / MMASS) * fmwy;
        const float thn = st[2] + PH * om + (PH / JINER) * fRm;
        const float ct2 = __cosf(thn), sn2 = __sinf(thn);
        const float fpwx = ct2 * fxp - sn2 * fyp;
        const float fpwy = sn2 * fxp + ct2 * fyp;
        st[0] = xn; st[1] = yn; st[2] = thn;
        st[3] = vx + (fmwx + fpwx) * (1.0f / MMASS);
        st[4] = vy + (fmwy + fpwy) * (1.0f / MMASS);
        st[5] = om + (fRm + fRp) * (1.0f / JINER);
        st[14] = u0n; st[15] = u1n; st[16] = u2n;

#pragma unroll
        for (int c = 0; c < D; ++c) outp[c] = st[c];

        ctrlp += stride;
        outp  += stride;
    }
}

extern "C" void kernel_launch(void* const* d_in, const int* in_sizes, int n_in,
                              void* d_out, int out_size, void* d_ws, size_t ws_size,
                              hipStream_t stream) {
    const float* traj = (const float*)d_in[0];
    const float* W1   = (const float*)d_in[1];
    const float* b1   = (const float*)d_in[2];
    const float* W2   = (const float*)d_in[3];
    const float* b2   = (const float*)d_in[4];
    float* out        = (float*)d_out;
    // 65536 rows / (4 waves * 32 rows) = 512 blocks of 128 threads (4 wave32s).
    PlanarPhysORD_wmma_kernel<<<BS / 128, 128, 0, stream>>>(traj, W1, b1, W2, b2, out);
}